// NAFLTEOURS_89189290869350
// MI455X (gfx1250) — compile-verified
//
#include <hip/hip_runtime.h>

#define WMAT 256
#define WP   257
#define NMAT 256
#define ITERS 10

typedef __attribute__((ext_vector_type(2))) float v2f;
typedef __attribute__((ext_vector_type(8))) float v8f;
typedef __attribute__((ext_vector_type(4))) unsigned int v4u;
typedef __attribute__((ext_vector_type(8))) int v8i;
typedef __attribute__((ext_vector_type(4))) int v4i;

__device__ __forceinline__ float wave_max32(float x) {
#pragma unroll
  for (int off = 16; off > 0; off >>= 1) x = fmaxf(x, __shfl_xor(x, off, 32));
  return x;
}
__device__ __forceinline__ float wave_sum32(float x) {
#pragma unroll
  for (int off = 16; off > 0; off >>= 1) x += __shfl_xor(x, off, 32);
  return x;
}

__global__ __launch_bounds__(256, 1)
void sinkhorn_ot_kernel(const float* __restrict__ attn,
                        const float* __restrict__ phi_p,
                        float* __restrict__ out) {
  // 257*257 floats = 264,196 B; + u/v: total ~266 KB < 320 KB WGP LDS (CDNA5 only)
  __shared__ float sim_s[WP * WP];
  __shared__ float u_s[WP];
  __shared__ float v_s[WP];

  const int tid  = threadIdx.x;
  const int lane = tid & 31;
  const int wv   = tid >> 5;   // wave 0..7
  const int hf   = lane >> 4;  // K-half within wave
  const int sub  = lane & 15;  // row/col within 16-group
  const int mat  = blockIdx.x; // 0..255 (n*H + h)

  const float phi      = phi_p[0];
  const float LOG_MU0  = -6.2383246250395075f; // log(1/(2W))
  const float LOG_MU_L = -0.6931471805599453f; // log(1/2)
  const float LOG_2W   =  6.2383246250395075f; // log(2W)
  const float NEG_HUGE = -3.402823466e+38f;

  // ---- TDM: async DMA 256x256 f32 tile HBM->LDS, pad 1 dword / 256 dwords
  // so rows land at LDS stride 257 (odd => conflict-free row AND col sweeps).
  if (wv == 0) {
    unsigned long long ga =
        (unsigned long long)(const void*)(attn + (size_t)mat * WMAT * WMAT);
    unsigned int lds_base = (unsigned int)(unsigned long long)(const void*)&sim_s[0];
    v4u g0;
    g0.x = 1u;                                             // count=1 valid user D#
    g0.y = lds_base;                                       // lds_addr
    g0.z = (unsigned int)(ga & 0xFFFFFFFFull);             // global_addr[31:0]
    g0.w = (unsigned int)((ga >> 32) & 0x1FFFFFFull) | (2u << 30); // [56:32] | type=2
    v8i g1;
    g1[0] = (int)((2u << 16) | (1u << 20) | (7u << 22));   // 4B elems, pad_en, ivl=256dw, amt=1dw
    g1[1] = (int)((unsigned)WMAT << 16);                   // tensor_dim0 lo16
    g1[2] = (int)((unsigned)WMAT << 16);                   // dim0 hi16=0 | tensor_dim1 lo16
    g1[3] = (int)((unsigned)WMAT << 16);                   // dim1 hi16=0 | tile_dim0=256
    g1[4] = (int)WMAT;                                     // tile_dim1=256, tile_dim2=0
    g1[5] = (int)WMAT;                                     // tensor_dim0_stride lo32
    g1[6] = 0;
    g1[7] = 0;
    v4i z4 = {0, 0, 0, 0};
    v8i z8 = {0, 0, 0, 0, 0, 0, 0, 0};
    __builtin_amdgcn_tensor_load_to_lds(g0, g1, z4, z4, z8, 0);
  }

  // Overlap with DMA: init u,v and the phi border (col 256 / row 256, untouched by TDM)
  for (int i = tid; i < WP; i += 256) {
    u_s[i] = 0.0f;
    v_s[i] = 0.0f;
    sim_s[i * WP + WMAT] = phi;
    sim_s[WMAT * WP + i] = phi;
  }
  if (wv == 0) __builtin_amdgcn_s_wait_tensorcnt(0);
  __syncthreads();

  const v2f ones2 = {1.0f, 1.0f};

  for (int it = 0; it < ITERS; ++it) {
    // ======== v-update: per-column logsumexp over rows of (sim + u) ========
#pragma unroll
    for (int g = 0; g < 2; ++g) {
      const int col = (wv + g * 8) * 16 + sub;
      float cm = NEG_HUGE;
      for (int i = hf; i < WP; i += 2)
        cm = fmaxf(cm, sim_s[i * WP + col] + u_s[i]);
      cm = fmaxf(cm, __shfl_xor(cm, 16, 32));
      // sum-of-exp via matrix pipe: D = ones(16x4) x E(4x16) -> column sums
      v8f acc = {0, 0, 0, 0, 0, 0, 0, 0};
      for (int c = 0; c < WMAT / 4; ++c) {
        const int i0 = 4 * c + 2 * hf;
        v2f b;
        b.x = __expf(sim_s[i0 * WP + col] + u_s[i0] - cm);
        b.y = __expf(sim_s[(i0 + 1) * WP + col] + u_s[i0 + 1] - cm);
        acc = __builtin_amdgcn_wmma_f32_16x16x4_f32(false, ones2, false, b,
                                                    (short)0, acc, false, false);
      }
      float s = acc[0];                                        // colsum(col), any M
      s += __expf(sim_s[WMAT * WP + col] + u_s[WMAT] - cm);    // tail row i=256
      if (hf == 0) v_s[col] = LOG_MU0 - (__logf(s) + cm);
    }
    if (wv == 0) { // border column j=256
      float m = NEG_HUGE;
      for (int i = lane; i < WP; i += 32)
        m = fmaxf(m, sim_s[i * WP + WMAT] + u_s[i]);
      m = wave_max32(m);
      float s = 0.0f;
      for (int i = lane; i < WP; i += 32)
        s += __expf(sim_s[i * WP + WMAT] + u_s[i] - m);
      s = wave_sum32(s);
      if (lane == 0) v_s[WMAT] = LOG_MU_L - (__logf(s) + m);
    }
    __syncthreads();

    // ======== u-update: per-row logsumexp over cols of (sim + v) ========
#pragma unroll
    for (int g = 0; g < 2; ++g) {
      const int rbase = (wv + g * 8) * 16;
      const int row = rbase + sub;
      float rm = NEG_HUGE;
      for (int j = hf; j < WP; j += 2)
        rm = fmaxf(rm, sim_s[row * WP + j] + v_s[j]);
      rm = fmaxf(rm, __shfl_xor(rm, 16, 32));
      // D = E(16x4) x ones(4x16) -> row sums; lanes carry K={2hf,2hf+1}
      v8f acc = {0, 0, 0, 0, 0, 0, 0, 0};
      for (int c = 0; c < WMAT / 4; ++c) {
        const int j0 = 4 * c + 2 * hf;
        v2f a;
        a.x = __expf(sim_s[row * WP + j0] + v_s[j0] - rm);
        a.y = __expf(sim_s[row * WP + j0 + 1] + v_s[j0 + 1] - rm);
        acc = __builtin_amdgcn_wmma_f32_16x16x4_f32(false, a, false, ones2,
                                                    (short)0, acc, false, false);
      }
      // lane 0 holds rowsums rows 0..7 in acc[0..7]; lane 16 rows 8..15
#pragma unroll
      for (int r = 0; r < 8; ++r) {
        const int rr = rbase + hf * 8 + r;
        float rmax_r = __shfl(rm, hf * 8 + r, 32);
        float s = acc[r];
        s += __expf(sim_s[rr * WP + WMAT] + v_s[WMAT] - rmax_r); // tail col j=256
        if (sub == 0) u_s[rr] = LOG_MU0 - (__logf(s) + rmax_r);
      }
    }
    if (wv == 0) { // border row i=256
      float m = NEG_HUGE;
      for (int j = lane; j < WP; j += 32)
        m = fmaxf(m, sim_s[WMAT * WP + j] + v_s[j]);
      m = wave_max32(m);
      float s = 0.0f;
      for (int j = lane; j < WP; j += 32)
        s += __expf(sim_s[WMAT * WP + j] + v_s[j] - m);
      s = wave_sum32(s);
      if (lane == 0) u_s[WMAT] = LOG_MU_L - (__logf(s) + m);
    }
    __syncthreads();
  }

  // ======== matched = exp(sim+u+v+log2W); argmax row, 3-tap window ========
  const int i = tid; // one row per thread, 0..255
  const float ui = u_s[i];
  float best = NEG_HUGE;
  int high = 0;
  for (int j = 0; j < WMAT; ++j) {
    float t = sim_s[i * WP + j] + v_s[j]; // exp is monotone: argmax preserved
    if (t > best) { best = t; high = j; }
  }
  float norm = 0.0f, wsum = 0.0f;
#pragma unroll
  for (int d = -1; d <= 1; ++d) {
    int jw = high + d;
    if (jw >= 0 && jw < WMAT) {
      float mv = __expf(sim_s[i * WP + jw] + ui + v_s[jw] + LOG_2W);
      float p = fmaxf((float)(i - jw), 0.0f);
      norm += mv;
      wsum += mv * p;
    }
  }
  float nc = (norm < 0.1f) ? 1.0f : norm;
  out[mat * WMAT + i] = wsum / nc;                  // disp_pred
  out[NMAT * WMAT + mat * WMAT + i] = 1.0f - nc;    // occ_pred
}

extern "C" void kernel_launch(void* const* d_in, const int* in_sizes, int n_in,
                              void* d_out, int out_size, void* d_ws, size_t ws_size,
                              hipStream_t stream) {
  const float* attn = (const float*)d_in[0];
  const float* phi  = (const float*)d_in[1];
  float* out = (float*)d_out;
  sinkhorn_ot_kernel<<<dim3(NMAT), dim3(256), 0, stream>>>(attn, phi, out);
  (void)in_sizes; (void)n_in; (void)out_size; (void)d_ws; (void)ws_size;
}